// Network_42992622633163
// MI455X (gfx1250) — compile-verified
//
#include <hip/hip_runtime.h>
#include <hip/hip_bf16.h>

#define BD 8
#define CD 128
#define CED 64
#define HD 64
#define WD 128
#define ND (HD * WD)   // 8192
#define CHB 6          // scan channels per block: 6*32KB + 64KB = 256KB LDS (<320KB WGP LDS)

typedef __attribute__((ext_vector_type(2))) float v2f;
typedef __attribute__((ext_vector_type(8))) float v8f;

#if defined(__has_builtin)
#if __has_builtin(__builtin_amdgcn_global_load_async_to_lds_b32) && \
    __has_builtin(__builtin_amdgcn_s_wait_asynccnt)
#define HAVE_ASYNC_LDS 1
#endif
#endif

#ifdef HAVE_ASYNC_LDS
typedef __attribute__((address_space(1))) int gint;  // global (AS1)
typedef __attribute__((address_space(3))) int lint;  // LDS (AS3)
#endif

// ---------------------------------------------------------------------------
// Kernel 1: embed GEMM   E[b][ce][n] = sum_c We[ce][c] * lf[b][c][n]
// M=64 (4 wave-tiles), N-tile=16 per block, K=128 via V_WMMA_F32_16X16X4_F32
// ---------------------------------------------------------------------------
__global__ __launch_bounds__(128) void k_embed(const float* __restrict__ lf,
                                               const float* __restrict__ We,
                                               float* __restrict__ E) {
  __shared__ float sW[CED][CD + 1];  // padded to kill bank conflicts
  __shared__ float sB[CD][17];
  const int b  = blockIdx.y;
  const int n0 = blockIdx.x << 4;
  const int t  = threadIdx.x;

  for (int i = t; i < CED * CD; i += 128) sW[i >> 7][i & 127] = We[i];
  const float* lfb = lf + (size_t)b * CD * ND + n0;
  for (int i = t; i < CD * 16; i += 128)
    sB[i >> 4][i & 15] = lfb[(size_t)(i >> 4) * ND + (i & 15)];
  __syncthreads();

  const int wid   = t >> 5, l = t & 31;
  const int m     = (wid << 4) + (l & 15);
  const int n     = l & 15;
  const int khalf = (l >> 4) << 1;  // lanes 0-15 -> K{0,1}, lanes 16-31 -> K{2,3}
  v8f acc = {};
  #pragma unroll 4
  for (int kk = 0; kk < CD; kk += 4) {
    v2f a, bb;
    a.x  = sW[m][kk + khalf];      a.y  = sW[m][kk + khalf + 1];
    bb.x = sB[kk + khalf][n];      bb.y = sB[kk + khalf + 1][n];
    acc = __builtin_amdgcn_wmma_f32_16x16x4_f32(false, a, false, bb,
                                                (short)0, acc, false, false);
  }
  float* Eb = E + (size_t)b * CED * ND + n0;
  const int mrow = (wid << 4) + ((l >> 4) << 3);  // C/D: VGPR r -> M=r / M=r+8
  #pragma unroll
  for (int r = 0; r < 8; ++r)
    Eb[(size_t)(mrow + r) * ND + n] = acc[r];
}

// ---------------------------------------------------------------------------
// Kernel 2: edge weights  w[k] = exp(-||E[:,order[k]] - E[:,order[pp[k]]]||^2)
// ---------------------------------------------------------------------------
__global__ __launch_bounds__(64) void k_weights(const float* __restrict__ E,
                                                const int* __restrict__ order,
                                                const int* __restrict__ parent,
                                                float* __restrict__ w) {
  const int b = blockIdx.y, k = blockIdx.x, t = threadIdx.x;  // t < 64
  const int n  = order[b * ND + k];
  const int np = order[b * ND + parent[b * ND + k]];
  const float* Eb = E + (size_t)b * CED * ND;
  const float d = Eb[(size_t)t * ND + n] - Eb[(size_t)t * ND + np];
  __shared__ float red[64];
  red[t] = d * d;
  __syncthreads();
  if (t == 0) {
    float s = 0.f;
    #pragma unroll
    for (int i = 0; i < 64; ++i) s += red[i];
    w[b * ND + k] = __expf(-s);
  }
}

// ---------------------------------------------------------------------------
// Kernel 3: LDS-resident tree scan (the CDNA5 320KB-LDS play).
// Each block owns CHB channel columns (32KB each) + packed (parent,weight)
// stream, entirely in LDS. One lane per channel walks the dependency chain at
// LDS latency instead of L2 latency. Gather from pixel order on the way in,
// write finished columns back coalesced (Y layout: [b][129][k]).
// ---------------------------------------------------------------------------
struct PW { int p; float w; };

__global__ __launch_bounds__(256) void k_scan(const float* __restrict__ lf,
                                              const float* __restrict__ w,
                                              const int* __restrict__ order,
                                              const int* __restrict__ parent,
                                              float* __restrict__ Y) {
  extern __shared__ float smem[];
  float* sA = smem;                        // CHB * ND floats
  PW*    sPW = (PW*)(smem + CHB * ND);     // ND entries, 8B each
  const int b   = blockIdx.y;
  const int cb  = blockIdx.x * CHB;
  const int t   = threadIdx.x;
  const int nch = min(CHB, (CD + 1) - cb);

  // pack parent + weight (one ds_load_b64 per scan step later)
  for (int k = t; k < ND; k += 256) {
    PW e; e.p = parent[b * ND + k]; e.w = w[b * ND + k];
    sPW[k] = e;
  }
  // gather this block's channel columns into BFS order (norm channel == 1)
  for (int k = t; k < ND; k += 256) {
    const int n = order[b * ND + k];
    for (int ch = 0; ch < nch; ++ch) {
      const int c = cb + ch;
      sA[ch * ND + k] = (c < CD) ? lf[((size_t)b * CD + c) * ND + n] : 1.0f;
    }
  }
  __syncthreads();

  if (t < nch) {
    float* Ac = sA + t * ND;
    // leaf -> root accumulate
    #pragma unroll 4
    for (int k = ND - 1; k >= 1; --k) {
      const PW e = sPW[k];
      Ac[e.p] += e.w * Ac[k];
    }
    // root -> leaf propagate (in place; parents p<k already final)
    #pragma unroll 4
    for (int k = 1; k < ND; ++k) {
      const PW e = sPW[k];
      const float av = Ac[k];
      Ac[k] = av + e.w * (Ac[e.p] - e.w * av);
    }
  }
  __syncthreads();

  // coalesced write-back of finished columns
  for (int ch = 0; ch < nch; ++ch)
    for (int k = t; k < ND; k += 256)
      Y[((size_t)b * (CD + 1) + cb + ch) * ND + k] = sA[ch * ND + k];
}

// ---------------------------------------------------------------------------
// Kernel 4: normalize + scatter back to pixel order + add latent
// ---------------------------------------------------------------------------
__global__ __launch_bounds__(128) void k_fuse(const float* __restrict__ Y,
                                              const float* __restrict__ latent,
                                              const int* __restrict__ order,
                                              float* __restrict__ fused) {
  const int b = blockIdx.y, k = blockIdx.x, c = threadIdx.x;
  const int n = order[b * ND + k];
  const float norm = Y[((size_t)b * (CD + 1) + CD) * ND + k];
  const float v = Y[((size_t)b * (CD + 1) + c) * ND + k] / norm;
  const size_t idx = ((size_t)b * CD + c) * ND + n;
  fused[idx] = latent[idx] + v;
}

// ---------------------------------------------------------------------------
// Kernel 5: refine GEMM   out[b][o][n] = sum_c Wr[o][c] * fused[b][c][n]
// M=128 (8 wave-tiles), K split in two 64-wide LDS slabs. B-tile staged via
// async global->LDS when the toolchain exposes it.
// ---------------------------------------------------------------------------
__global__ __launch_bounds__(256) void k_refine(const float* __restrict__ fused,
                                                const float* __restrict__ Wr,
                                                float* __restrict__ out) {
  __shared__ float sW[CD][65];
  __shared__ float sB[64][17];
  const int b  = blockIdx.y;
  const int n0 = blockIdx.x << 4;
  const int t  = threadIdx.x;
  const int wid   = t >> 5, l = t & 31;
  const int m     = (wid << 4) + (l & 15);
  const int n     = l & 15;
  const int khalf = (l >> 4) << 1;
  v8f acc = {};
  const float* fb = fused + (size_t)b * CD * ND + n0;

  for (int kh = 0; kh < 2; ++kh) {
    const int kbase = kh << 6;
    __syncthreads();
    for (int i = t; i < CD * 64; i += 256)
      sW[i >> 6][i & 63] = Wr[(size_t)(i >> 6) * CD + kbase + (i & 63)];
#ifdef HAVE_ASYNC_LDS
    for (int i = t; i < 64 * 16; i += 256) {
      const float* gp = &fb[(size_t)(kbase + (i >> 4)) * ND + (i & 15)];
      __builtin_amdgcn_global_load_async_to_lds_b32(
          (gint*)(unsigned long long)gp,
          (lint*)(unsigned long long)&sB[i >> 4][i & 15], 0, 0);
    }
    __builtin_amdgcn_s_wait_asynccnt(0);
#else
    for (int i = t; i < 64 * 16; i += 256)
      sB[i >> 4][i & 15] = fb[(size_t)(kbase + (i >> 4)) * ND + (i & 15)];
#endif
    __syncthreads();
    #pragma unroll 4
    for (int kk = 0; kk < 64; kk += 4) {
      v2f a, bb;
      a.x  = sW[m][kk + khalf];      a.y  = sW[m][kk + khalf + 1];
      bb.x = sB[kk + khalf][n];      bb.y = sB[kk + khalf + 1][n];
      acc = __builtin_amdgcn_wmma_f32_16x16x4_f32(false, a, false, bb,
                                                  (short)0, acc, false, false);
    }
  }
  float* ob = out + (size_t)b * CD * ND + n0;
  const int mrow = (wid << 4) + ((l >> 4) << 3);
  #pragma unroll
  for (int r = 0; r < 8; ++r)
    ob[(size_t)(mrow + r) * ND + n] = acc[r];
}

// ---------------------------------------------------------------------------
extern "C" void kernel_launch(void* const* d_in, const int* in_sizes, int n_in,
                              void* d_out, int out_size, void* d_ws, size_t ws_size,
                              hipStream_t stream) {
  const float* latent = (const float*)d_in[0];
  const float* lf     = (const float*)d_in[1];
  const float* We     = (const float*)d_in[2];
  const float* Wr     = (const float*)d_in[3];
  const int*   order  = (const int*)d_in[4];
  const int*   parent = (const int*)d_in[5];
  float* out = (float*)d_out;

  // workspace (floats): E | w | Y | fused   (~84 MB, L2-resident)
  float* E     = (float*)d_ws;
  float* wbuf  = E + (size_t)BD * CED * ND;
  float* Ybuf  = wbuf + (size_t)BD * ND;
  float* fused = Ybuf + (size_t)BD * (CD + 1) * ND;

  const size_t scan_lds = ((size_t)CHB * ND + 2 * ND) * sizeof(float);  // 256 KB
  const int    nchb     = (CD + 1 + CHB - 1) / CHB;                     // 22

  k_embed  <<<dim3(ND / 16, BD), 128, 0,        stream>>>(lf, We, E);
  k_weights<<<dim3(ND, BD),      64,  0,        stream>>>(E, order, parent, wbuf);
  k_scan   <<<dim3(nchb, BD),    256, scan_lds, stream>>>(lf, wbuf, order, parent, Ybuf);
  k_fuse   <<<dim3(ND, BD),      128, 0,        stream>>>(Ybuf, latent, order, fused);
  k_refine <<<dim3(ND / 16, BD), 256, 0,        stream>>>(fused, Wr, out);
}